// LINKX_12962211299588
// MI455X (gfx1250) — compile-verified
//
#include <hip/hip_runtime.h>

// ---------------- problem constants (match reference) ----------------
#define N_NODES 100000
#define EDGES   1600000
#define HCH     128            // hidden = in = out channels
#define LDSS    132            // A-tile row stride (floats): conflict-free A-frag reads
#define PSTR    288            // packed weight pair-row stride (dwords): PSTR%64==32 -> conflict-free B-frag reads

typedef __attribute__((ext_vector_type(2))) float v2f;
typedef __attribute__((ext_vector_type(4))) float v4f;
typedef __attribute__((ext_vector_type(8))) float v8f;

#if defined(__HIP_DEVICE_COMPILE__)
#if !__has_builtin(__builtin_amdgcn_wmma_f32_16x16x4_f32)
#error "missing __builtin_amdgcn_wmma_f32_16x16x4_f32 on gfx1250 device pass"
#endif
#endif

// D = A(16x4 f32) x B(4x16 f32) + C(16x16 f32), full fp32 precision
__device__ __forceinline__ v8f wmma4(v2f a, v2f b, v8f c) {
#if defined(__HIP_DEVICE_COMPILE__)
    return __builtin_amdgcn_wmma_f32_16x16x4_f32(
        /*neg_a=*/false, a, /*neg_b=*/false, b,
        /*c_mod=*/(short)0, c, /*reuse_a=*/false, /*reuse_b=*/false);
#else
    return c;
#endif
}

// ---------------- kernel 0: zero the SpMM accumulator ----------------
__global__ void linkx_zero(v4f* __restrict__ p, int n4) {
    int i = blockIdx.x * blockDim.x + threadIdx.x;
    if (i < n4) { v4f z = {0.f, 0.f, 0.f, 0.f}; p[i] = z; }
}

// ---------------- kernel 1: SpMM scatter (one wave per edge) ---------
__global__ void linkx_scatter(const float* __restrict__ W_edge,
                              const long long* __restrict__ ei,   // [2, E] int64
                              const float* __restrict__ ew,
                              float* __restrict__ acc) {
    int gid = blockIdx.x * blockDim.x + threadIdx.x;
    int e = gid >> 5;
    if (e >= EDGES) return;
    int c = (gid & 31) << 2;                 // channel base: 0..124
    int s = (int)ei[e];
    int d = (int)ei[EDGES + e];
    float w = ew[e];
    v4f v = *(const v4f*)(W_edge + s * HCH + c);
    float* dst = acc + d * HCH + c;
    __hip_atomic_fetch_add(dst + 0, w * v.x, __ATOMIC_RELAXED, __HIP_MEMORY_SCOPE_AGENT);
    __hip_atomic_fetch_add(dst + 1, w * v.y, __ATOMIC_RELAXED, __HIP_MEMORY_SCOPE_AGENT);
    __hip_atomic_fetch_add(dst + 2, w * v.z, __ATOMIC_RELAXED, __HIP_MEMORY_SCOPE_AGENT);
    __hip_atomic_fetch_add(dst + 3, w * v.w, __ATOMIC_RELAXED, __HIP_MEMORY_SCOPE_AGENT);
}

// ---------------- kernel 2: fused GEMM chain, WMMA fp32 --------------
// 4 waves/block; each wave owns a 16-row tile + private A LDS buffer.
// Weights staged once per block into LDS in a PAIR-INTERLEAVED layout:
//   ldsW[p*PSTR + 2n + {0,1}] = { W[2p][n], W[2p+1][n] }
// so each B-fragment is a single aligned ds_load_b64 into a VGPR pair.
__launch_bounds__(128)
__global__ void linkx_fused(const float* __restrict__ x,
                            const float* __restrict__ acc,
                            const float* __restrict__ b_edge,
                            const float* __restrict__ W_node, const float* __restrict__ b_node,
                            const float* __restrict__ W_cat1, const float* __restrict__ b_cat1,
                            const float* __restrict__ W_cat2, const float* __restrict__ b_cat2,
                            const float* __restrict__ W_final, const float* __restrict__ b_final,
                            float* __restrict__ out) {
    __shared__ float ldsA[4 * 16 * LDSS];    // per-wave activation tiles (33 KB)
    __shared__ float ldsW[64 * PSTR];        // shared packed weight matrix (72 KB)

    const int tid  = threadIdx.x;
    const int lane = tid & 31;
    const int wave = tid >> 5;
    const int tile = blockIdx.x * 4 + wave;
    const bool active = (tile * 16 < N_NODES);   // wave-uniform predicate
    const int rowBase = tile * 16;

    float* A = ldsA + wave * 16 * LDSS;

    const int mrow = lane & 15;                // A row / B col / C col within tile
    const int kq   = (lane >> 4) << 1;         // 0 or 2 : K sub-offset per A/B layout
    const int cm   = (lane >> 4) << 3;         // 0 or 8 : M offset for C fragments

    // cooperative pair-interleaved copy of one 128x128 weight matrix into ldsW
    auto copyW = [&](const float* __restrict__ W) {
#pragma unroll 2
        for (int it = 0; it < 16; ++it) {
            int idx = it * 128 + tid;          // 0..2047 (64 pair-rows x 32 col-groups)
            int p   = idx >> 5;                // pair-row (k = 2p, 2p+1)
            int cg  = (idx & 31) << 2;         // col base 0..124
            v4f r0 = *(const v4f*)(W + (2 * p)     * HCH + cg);
            v4f r1 = *(const v4f*)(W + (2 * p + 1) * HCH + cg);
            float* dstp = ldsW + p * PSTR + 2 * cg;
            v2f t0 = {r0.x, r1.x}; *(v2f*)(dstp + 0) = t0;
            v2f t1 = {r0.y, r1.y}; *(v2f*)(dstp + 2) = t1;
            v2f t2 = {r0.z, r1.z}; *(v2f*)(dstp + 4) = t2;
            v2f t3 = {r0.w, r1.w}; *(v2f*)(dstp + 6) = t3;
        }
    };

    // GEMM over A(LDS 16x128) x W(LDS packed), accumulating into c[8]
    auto gemm = [&](v8f* c) {
        for (int kb = 0; kb < HCH; kb += 4) {
            v2f a = *(const v2f*)(A + mrow * LDSS + kb + kq);
            const float* Wp = ldsW + ((kb + kq) >> 1) * PSTR;   // pair-row base
#pragma unroll
            for (int j = 0; j < 8; ++j) {
                v2f b = *(const v2f*)(Wp + 2 * (j * 16 + mrow));  // one ds_load_b64
                c[j] = wmma4(a, b, c[j]);
            }
        }
    };

    // stage a 16x128 activation tile into the wave's private A buffer
    auto stage = [&](const float* __restrict__ src, const float* __restrict__ bias) {
        int c = lane * 4;
        v4f bb = {0.f, 0.f, 0.f, 0.f};
        if (bias) bb = *(const v4f*)(bias + c);
#pragma unroll 4
        for (int r = 0; r < 16; ++r) {
            v4f v = *(const v4f*)(src + (size_t)(rowBase + r) * HCH + c);
            v4f s = {v.x + bb.x, v.y + bb.y, v.z + bb.z, v.w + bb.w};
            *(v4f*)(A + r * LDSS + c) = s;
        }
    };

    v8f c1[8], c2[8];

    // ---- phase 1: agg = acc + b_edge ; t1 = agg + agg@W_cat1 + b_cat1 ----
    copyW(W_cat1);
    if (active) {
        stage(acc, b_edge);
#pragma unroll
        for (int j = 0; j < 8; ++j) {
            float bc = b_cat1[j * 16 + mrow];
#pragma unroll
            for (int r = 0; r < 8; ++r)
                c1[j][r] = A[(r + cm) * LDSS + j * 16 + mrow] + bc;   // C init = agg + b_cat1
        }
    }
    __syncthreads();
    if (active) gemm(c1);                                             // c1 = t1
    __syncthreads();

    // ---- phase 2: xh = x @ W_node + b_node ----
    copyW(W_node);
    if (active) {
        stage(x, nullptr);
#pragma unroll
        for (int j = 0; j < 8; ++j) {
            float bn = b_node[j * 16 + mrow];
#pragma unroll
            for (int r = 0; r < 8; ++r) c2[j][r] = bn;                // C init = b_node
        }
    }
    __syncthreads();
    if (active) gemm(c2);                                             // c2 = xh
    __syncthreads();

    // ---- phase 3: t2 = t1 + xh + xh@W_cat2 + b_cat2 ----
    copyW(W_cat2);
    if (active) {
#pragma unroll
        for (int j = 0; j < 8; ++j) {
            float bc = b_cat2[j * 16 + mrow];
#pragma unroll
            for (int r = 0; r < 8; ++r) {
                A[(r + cm) * LDSS + j * 16 + mrow] = c2[j][r];        // xh -> A (GEMM3 input)
                c1[j][r] += c2[j][r] + bc;                            // C init = t1 + xh + b_cat2
            }
        }
    }
    __syncthreads();
    if (active) gemm(c1);                                             // c1 = t2
    __syncthreads();

    // ---- phase 4: out = relu(t2) @ W_final + b_final ----
    copyW(W_final);
    if (active) {
#pragma unroll
        for (int j = 0; j < 8; ++j) {
            float bf = b_final[j * 16 + mrow];
#pragma unroll
            for (int r = 0; r < 8; ++r) {
                float v = c1[j][r];
                A[(r + cm) * LDSS + j * 16 + mrow] = v > 0.f ? v : 0.f;  // relu(t2) -> A
                c2[j][r] = bf;                                           // C init = b_final
            }
        }
    }
    __syncthreads();
    if (active) {
        gemm(c2);
        // write C fragments: lanes 0-15 cover 64B-contiguous runs per (j,r)
#pragma unroll
        for (int j = 0; j < 8; ++j)
#pragma unroll
            for (int r = 0; r < 8; ++r)
                out[(size_t)(rowBase + r + cm) * HCH + j * 16 + mrow] = c2[j][r];
    }
}

// ---------------- host entry ----------------
extern "C" void kernel_launch(void* const* d_in, const int* in_sizes, int n_in,
                              void* d_out, int out_size, void* d_ws, size_t ws_size,
                              hipStream_t stream) {
    const float*     x       = (const float*)d_in[0];
    const long long* ei      = (const long long*)d_in[1];
    const float*     ew      = (const float*)d_in[2];
    const float*     W_edge  = (const float*)d_in[3];
    const float*     b_edge  = (const float*)d_in[4];
    const float*     W_node  = (const float*)d_in[5];
    const float*     b_node  = (const float*)d_in[6];
    const float*     W_cat1  = (const float*)d_in[7];
    const float*     b_cat1  = (const float*)d_in[8];
    const float*     W_cat2  = (const float*)d_in[9];
    const float*     b_cat2  = (const float*)d_in[10];
    const float*     W_final = (const float*)d_in[11];
    const float*     b_final = (const float*)d_in[12];
    float* out = (float*)d_out;

    const size_t accBytes = (size_t)N_NODES * HCH * sizeof(float);
    float* acc = (ws_size >= accBytes) ? (float*)d_ws : out;  // out-fallback safe (read-before-write per tile)

    // 1) zero accumulator
    int n4 = N_NODES * HCH / 4;
    linkx_zero<<<(n4 + 255) / 256, 256, 0, stream>>>((v4f*)acc, n4);

    // 2) edge scatter: one wave (32 lanes x 4ch) per edge
    long long sthreads = (long long)EDGES * 32;
    linkx_scatter<<<(int)((sthreads + 255) / 256), 256, 0, stream>>>(W_edge, ei, ew, acc);

    // 3) fused GEMM chain: 6250 row-tiles, 4 waves/block
    int tiles = (N_NODES + 15) / 16;
    linkx_fused<<<(tiles + 3) / 4, 128, 0, stream>>>(x, acc, b_edge,
                                                     W_node, b_node,
                                                     W_cat1, b_cat1,
                                                     W_cat2, b_cat2,
                                                     W_final, b_final, out);
}